// TransitionDown_32049045963335
// MI455X (gfx1250) — compile-verified
//
#include <hip/hip_runtime.h>
#include <hip/hip_bf16.h>

#define BATCHES 16
#define PPC     4096      // points per cloud
#define SPC     1024      // fps samples per cloud
#define KNN_K   16
#define IN_F    256
#define OUT_F   512
#define NTOT    (BATCHES * PPC)   // 65536
#define NQ      (BATCHES * SPC)   // 16384

typedef float v2f __attribute__((ext_vector_type(2)));
typedef float v8f __attribute__((ext_vector_type(8)));

// ---------------------------------------------------------------------------
// 1) Farthest point sampling: one block (1024 threads) per cloud, 4 pts/thread
// ---------------------------------------------------------------------------
__global__ void fps_kernel(const float* __restrict__ pos, int* __restrict__ fps) {
    const int b = blockIdx.x;
    const float* p = pos + (size_t)b * PPC * 3;
    const int t = threadIdx.x;            // 0..1023

    float mind[4];
#pragma unroll
    for (int i = 0; i < 4; i++) mind[i] = 3.4e38f;

    __shared__ float s_val[1024];
    __shared__ int   s_idx[1024];
    __shared__ float s_last[3];

    int last = 0;
    for (int s = 0; s < SPC; s++) {
        if (t == 0) fps[b * SPC + s] = last;
        if (t < 3)  s_last[t] = p[last * 3 + t];
        __syncthreads();
        const float lx = s_last[0], ly = s_last[1], lz = s_last[2];
        float best = -1.0f; int bi = 0;
#pragma unroll
        for (int i = 0; i < 4; i++) {
            const int pt = t + i * 1024;
            const float dx = p[pt * 3 + 0] - lx;
            const float dy = p[pt * 3 + 1] - ly;
            const float dz = p[pt * 3 + 2] - lz;
            const float d = dx * dx + dy * dy + dz * dz;
            mind[i] = fminf(mind[i], d);
            if (mind[i] > best) { best = mind[i]; bi = pt; }  // strict > keeps lowest idx
        }
        s_val[t] = best; s_idx[t] = bi;
        __syncthreads();
        for (int off = 512; off > 0; off >>= 1) {
            if (t < off) {
                const float v2 = s_val[t + off]; const int i2 = s_idx[t + off];
                if (v2 > s_val[t] || (v2 == s_val[t] && i2 < s_idx[t])) {
                    s_val[t] = v2; s_idx[t] = i2;
                }
            }
            __syncthreads();
        }
        last = s_idx[0];
        __syncthreads();
    }
}

// ---------------------------------------------------------------------------
// 2) |x|^2 per point (feature space)
// ---------------------------------------------------------------------------
__global__ void xnorm_kernel(const float* __restrict__ f, float* __restrict__ xn) {
    const int pidx = blockIdx.x * blockDim.x + threadIdx.x; // 0..65535
    const float* r = f + (size_t)pidx * IN_F;
    float s = 0.f;
    for (int k = 0; k < IN_F; k++) { const float v = r[k]; s += v * v; }
    xn[pidx] = s;
}

// ---------------------------------------------------------------------------
// 3) H = X @ W + bias via v_wmma_f32_16x16x4_f32.
//    Block = 8 waves; block tile 128Mx64N; wave = 16Mx64N (4 accum tiles).
// ---------------------------------------------------------------------------
__global__ void gemm_bias_kernel(const float* __restrict__ A, const float* __restrict__ Bw,
                                 const float* __restrict__ bias, float* __restrict__ H) {
    const int wave = threadIdx.x >> 5;             // 0..7
    const int lane = threadIdx.x & 31;
    const int hl   = lane & 15;
    const int ksel = (lane >> 4) * 2;              // 0 or 2
    const int m0   = blockIdx.x * 128 + wave * 16;
    const int n0   = blockIdx.y * 64;

    v8f acc0 = {}, acc1 = {}, acc2 = {}, acc3 = {};
    const float* arow = A + (size_t)(m0 + hl) * IN_F;

    for (int k = 0; k < IN_F; k += 4) {
        const v2f a = *(const v2f*)(arow + k + ksel);
        v2f bf;
        const int c0 = n0 + hl;
        bf.x = Bw[(k + ksel) * OUT_F + c0];      bf.y = Bw[(k + ksel + 1) * OUT_F + c0];
        acc0 = __builtin_amdgcn_wmma_f32_16x16x4_f32(false, a, false, bf, (short)0, acc0, false, false);
        bf.x = Bw[(k + ksel) * OUT_F + c0 + 16]; bf.y = Bw[(k + ksel + 1) * OUT_F + c0 + 16];
        acc1 = __builtin_amdgcn_wmma_f32_16x16x4_f32(false, a, false, bf, (short)0, acc1, false, false);
        bf.x = Bw[(k + ksel) * OUT_F + c0 + 32]; bf.y = Bw[(k + ksel + 1) * OUT_F + c0 + 32];
        acc2 = __builtin_amdgcn_wmma_f32_16x16x4_f32(false, a, false, bf, (short)0, acc2, false, false);
        bf.x = Bw[(k + ksel) * OUT_F + c0 + 48]; bf.y = Bw[(k + ksel + 1) * OUT_F + c0 + 48];
        acc3 = __builtin_amdgcn_wmma_f32_16x16x4_f32(false, a, false, bf, (short)0, acc3, false, false);
    }

    const int rbase = m0 + ((lane >> 4) ? 8 : 0);
    v8f accs[4] = {acc0, acc1, acc2, acc3};
#pragma unroll
    for (int nt = 0; nt < 4; nt++) {
        const int col = n0 + nt * 16 + hl;
        const float bv = bias[col];
#pragma unroll
        for (int r = 0; r < 8; r++)
            H[(size_t)(rbase + r) * OUT_F + col] = accs[nt][r] + bv;
    }
}

// ---------------------------------------------------------------------------
// 4) Column statistics: 128 blocks x 512 threads (1 channel/thread, 512 rows/block)
// ---------------------------------------------------------------------------
__global__ void colstat_kernel(const float* __restrict__ H, float* __restrict__ part) {
    const int c = threadIdx.x;            // 0..511
    const int r0 = blockIdx.x * 512;      // 128 blocks
    float s = 0.f, s2 = 0.f;
    for (int r = 0; r < 512; r++) {
        const float v = H[(size_t)(r0 + r) * OUT_F + c];
        s += v; s2 += v * v;
    }
    part[blockIdx.x * OUT_F + c] = s;
    part[128 * OUT_F + blockIdx.x * OUT_F + c] = s2;
}

__global__ void finalize_stats(const float* __restrict__ part,
                               float* __restrict__ mu, float* __restrict__ rsig) {
    const int c = threadIdx.x; // 512
    float s = 0.f, s2 = 0.f;
    for (int bi = 0; bi < 128; bi++) {
        s  += part[bi * OUT_F + c];
        s2 += part[128 * OUT_F + bi * OUT_F + c];
    }
    const float m = s / (float)NTOT;
    const float v = s2 / (float)NTOT - m * m;
    mu[c]   = m;
    rsig[c] = rsqrtf(v + 1e-5f);
}

// ---------------------------------------------------------------------------
// 5) In-place BN + ReLU on H (float4 vectorized, OUT_F divisible by 4)
// ---------------------------------------------------------------------------
__global__ void bn_relu_kernel(float* __restrict__ H, const float* __restrict__ mu,
                               const float* __restrict__ rsig, const float* __restrict__ gamma,
                               const float* __restrict__ beta) {
    const size_t i = (size_t)blockIdx.x * blockDim.x + threadIdx.x;
    const size_t e = i * 4;
    const int c = (int)(e & (OUT_F - 1));
    float4 v = *(float4*)(H + e);
    v.x = fmaxf(gamma[c + 0] * (v.x - mu[c + 0]) * rsig[c + 0] + beta[c + 0], 0.f);
    v.y = fmaxf(gamma[c + 1] * (v.y - mu[c + 1]) * rsig[c + 1] + beta[c + 1], 0.f);
    v.z = fmaxf(gamma[c + 2] * (v.z - mu[c + 2]) * rsig[c + 2] + beta[c + 2], 0.f);
    v.w = fmaxf(gamma[c + 3] * (v.w - mu[c + 3]) * rsig[c + 3] + beta[c + 3], 0.f);
    *(float4*)(H + e) = v;
}

// ---------------------------------------------------------------------------
// 6) Fused KNN: per block = 16 queries of one cloud. 8 waves compute a 16x128
//    WMMA strip of scores s = -2*q.x + |x|^2 into LDS; 16 threads maintain
//    per-query sorted top-16 (index-ordered, strict <  ==  top_k tie-break).
// ---------------------------------------------------------------------------
__global__ void knn_kernel(const float* __restrict__ F, const int* __restrict__ fps,
                           const float* __restrict__ xn, int* __restrict__ knn) {
    const int blk  = blockIdx.x;          // 1024 = 16 clouds * 64 tiles
    const int b    = blk >> 6;
    const int mt   = blk & 63;
    const int wave = threadIdx.x >> 5;
    const int lane = threadIdx.x & 31;
    const int hl   = lane & 15;
    const int ksel = (lane >> 4) * 2;

    __shared__ float sdist[16][129];
    __shared__ int   sq[16];
    if (threadIdx.x < 16) sq[threadIdx.x] = fps[b * SPC + mt * 16 + threadIdx.x];
    __syncthreads();

    const float* qrow = F + ((size_t)b * PPC + sq[hl]) * IN_F;

    float kd[KNN_K]; int ki[KNN_K];
#pragma unroll
    for (int j = 0; j < KNN_K; j++) { kd[j] = 3.4e38f; ki[j] = PPC; }

    for (int n0 = 0; n0 < PPC; n0 += 128) {
        const int col = n0 + wave * 16 + hl;
        const float* xrow = F + ((size_t)b * PPC + col) * IN_F;
        v8f acc = {};
        for (int k = 0; k < IN_F; k += 4) {
            const v2f a  = *(const v2f*)(qrow + k + ksel);
            const v2f bb = *(const v2f*)(xrow + k + ksel);   // B = x^T: contiguous in k
            acc = __builtin_amdgcn_wmma_f32_16x16x4_f32(false, a, false, bb, (short)0, acc, false, false);
        }
        const float xv = xn[(size_t)b * PPC + col];
        const int rofs = (lane >> 4) ? 8 : 0;
#pragma unroll
        for (int r = 0; r < 8; r++)
            sdist[r + rofs][wave * 16 + hl] = -2.f * acc[r] + xv;
        __syncthreads();

        if (threadIdx.x < 16) {
            const int q = threadIdx.x;
            for (int j = 0; j < 128; j++) {
                const float d = sdist[q][j];
                if (d < kd[KNN_K - 1]) {
                    const int idx = n0 + j;
                    int p = KNN_K - 1;
                    while (p > 0 && kd[p - 1] > d) { kd[p] = kd[p - 1]; ki[p] = ki[p - 1]; p--; }
                    kd[p] = d; ki[p] = idx;
                }
            }
        }
        __syncthreads();
    }

    if (threadIdx.x < 16) {
        const size_t base = ((size_t)b * SPC + mt * 16 + threadIdx.x) * KNN_K;
        for (int j = 0; j < KNN_K; j++) knn[base + j] = ki[j];
    }
}

// ---------------------------------------------------------------------------
// 7) Max-pool over K neighbors: one block per query, 2 channels/thread
// ---------------------------------------------------------------------------
__global__ void maxpool_kernel(const float* __restrict__ H, const int* __restrict__ knn,
                               float* __restrict__ out) {
    const int qi = blockIdx.x;          // 16384
    const int b  = qi >> 10;
    const int c  = threadIdx.x;         // 256
    __shared__ int sidx[KNN_K];
    if (threadIdx.x < KNN_K) sidx[threadIdx.x] = knn[(size_t)qi * KNN_K + threadIdx.x];
    __syncthreads();
    float m0 = -3.4e38f, m1 = -3.4e38f;
    for (int j = 0; j < KNN_K; j++) {
        const float* row = H + ((size_t)b * PPC + sidx[j]) * OUT_F;
        m0 = fmaxf(m0, row[c]);
        m1 = fmaxf(m1, row[c + 256]);
    }
    out[(size_t)qi * OUT_F + c]       = m0;
    out[(size_t)qi * OUT_F + c + 256] = m1;
}

// ---------------------------------------------------------------------------
// 8) Gathered positions + batch ids
// ---------------------------------------------------------------------------
__global__ void aux_kernel(const float* __restrict__ pos, const int* __restrict__ batch,
                           const int* __restrict__ fps, float* __restrict__ out_pos,
                           int* __restrict__ out_batch) {
    const int qi = blockIdx.x * blockDim.x + threadIdx.x; // 16384
    const int b  = qi >> 10;
    const int g  = b * PPC + fps[qi];
    out_pos[qi * 3 + 0] = pos[g * 3 + 0];
    out_pos[qi * 3 + 1] = pos[g * 3 + 1];
    out_pos[qi * 3 + 2] = pos[g * 3 + 2];
    out_batch[qi] = batch[g];
}

// ---------------------------------------------------------------------------
extern "C" void kernel_launch(void* const* d_in, const int* in_sizes, int n_in,
                              void* d_out, int out_size, void* d_ws, size_t ws_size,
                              hipStream_t stream) {
    const float* features = (const float*)d_in[0];   // [65536,256]
    const float* positions = (const float*)d_in[1];  // [65536,3]
    const int*   batch     = (const int*)d_in[2];    // [65536]
    const float* W         = (const float*)d_in[3];  // [256,512]
    const float* bias      = (const float*)d_in[4];  // [512]
    const float* gamma     = (const float*)d_in[5];  // [512]
    const float* beta      = (const float*)d_in[6];  // [512]

    char* ws = (char*)d_ws;
    float* h = (float*)ws;                                   size_t off = (size_t)NTOT * OUT_F * 4;
    int*   fps  = (int*)(ws + off);                          off += (size_t)NQ * 4;
    int*   knn  = (int*)(ws + off);                          off += (size_t)NQ * KNN_K * 4;
    float* xn   = (float*)(ws + off);                        off += (size_t)NTOT * 4;
    float* part = (float*)(ws + off);                        off += (size_t)256 * OUT_F * 4;
    float* mu   = (float*)(ws + off);                        off += (size_t)OUT_F * 4;
    float* rsig = (float*)(ws + off);                        off += (size_t)OUT_F * 4;

    float* out_feat  = (float*)d_out;                        // [16384,512]
    float* out_pos   = out_feat + (size_t)NQ * OUT_F;        // [16384,3]
    int*   out_batch = (int*)(out_pos + (size_t)NQ * 3);     // [16384]

    fps_kernel<<<BATCHES, 1024, 0, stream>>>(positions, fps);
    xnorm_kernel<<<NTOT / 256, 256, 0, stream>>>(features, xn);
    gemm_bias_kernel<<<dim3(NTOT / 128, OUT_F / 64), 256, 0, stream>>>(features, W, bias, h);
    colstat_kernel<<<128, 512, 0, stream>>>(h, part);
    finalize_stats<<<1, 512, 0, stream>>>(part, mu, rsig);
    bn_relu_kernel<<<((size_t)NTOT * OUT_F / 4) / 256, 256, 0, stream>>>(h, mu, rsig, gamma, beta);
    knn_kernel<<<BATCHES * (SPC / 16), 256, 0, stream>>>(features, fps, xn, knn);
    maxpool_kernel<<<NQ, 256, 0, stream>>>(h, knn, out_feat);
    aux_kernel<<<NQ / 256, 256, 0, stream>>>(positions, batch, fps, out_pos, out_batch);
}